// EQUICAT_1271310320428
// MI455X (gfx1250) — compile-verified
//
#include <hip/hip_runtime.h>
#include <math.h>

typedef float v2f __attribute__((ext_vector_type(2)));
typedef float v8f __attribute__((ext_vector_type(8)));

#define NN 20000
#define EE 320000
#define CC 64
#define ZZ 10
#define RMAXF 5.0f
#define AVGF 16.0f
#define PI_F 3.14159265358979f

__device__ __forceinline__ float silu_fast(float v) {
  // v / (1 + exp(-v)) with v_rcp_f32 instead of IEEE divide
  return v * __builtin_amdgcn_rcpf(1.f + __expf(-v));
}

// ---------------- Hz = (W_embed @ W_up) / (sqrt(Z)*sqrt(C)) : [10,64] -------
__global__ void hz_kernel(const float* __restrict__ We,
                          const float* __restrict__ Wu,
                          float* __restrict__ Hz) {
  int z = blockIdx.x, c = threadIdx.x;
  float s = 0.f;
  #pragma unroll 8
  for (int k = 0; k < CC; ++k) s += We[z * CC + k] * Wu[k * CC + c];
  Hz[z * CC + c] = s * (0.316227766f * 0.125f); // 1/sqrt(10) * 1/8
}

// ---------------- zero the message accumulator ------------------------------
__global__ void zero_kernel(float4* __restrict__ p, int n4) {
  int i = blockIdx.x * blockDim.x + threadIdx.x;
  if (i < n4) p[i] = make_float4(0.f, 0.f, 0.f, 0.f);
}

// ---------------- fused edge kernel: geometry + MLP (WMMA) + scatter --------
// 256 threads = 8 wave32; 32 edges per block.
__global__ __launch_bounds__(256) void edge_kernel(
    const float* __restrict__ pos, const int* __restrict__ spec,
    const int* __restrict__ eidx, const float* __restrict__ Hz,
    const float* __restrict__ w1, const float* __restrict__ w2,
    const float* __restrict__ w3, const float* __restrict__ w4,
    float* __restrict__ msg) {
  __shared__ float sF[32 * 12];   // edge features [32,8] (ld 12)
  __shared__ float sY[32 * 12];   // sph harm [32,9] (ld 12)
  __shared__ int   sS[32], sR[32];
  __shared__ float sH1[32 * 68];  // hidden ping (ld 68: bank-conflict-free)
  __shared__ float sH2[32 * 68];  // hidden pong
  __shared__ float sT[32 * 196];  // tpw [32,192] (ld 196)

  const int tid = threadIdx.x;
  const int e0  = blockIdx.x * 32;

  // ---- per-edge geometry / radial basis (threads 0..31) ----
  if (tid < 32) {
    int e = e0 + tid;
    int s = eidx[e];
    int r = eidx[EE + e];
    sS[tid] = s; sR[tid] = r;
    float dx = pos[r * 3 + 0] - pos[s * 3 + 0];
    float dy = pos[r * 3 + 1] - pos[s * 3 + 1];
    float dz = pos[r * 3 + 2] - pos[s * 3 + 2];
    float rr  = sqrtf(dx * dx + dy * dy + dz * dz + 1e-12f);
    float inv = __builtin_amdgcn_rcpf(rr);
    float x = dx * inv, y = dy * inv, z = dz * inv;
    const float s3 = 1.73205081f, s5 = 2.23606798f, s15 = 3.87298335f;
    float* Yp = &sY[tid * 12];
    Yp[0] = 1.f;          Yp[1] = s3 * x;       Yp[2] = s3 * y;
    Yp[3] = s3 * z;       Yp[4] = s15 * x * y;  Yp[5] = s15 * y * z;
    Yp[6] = 0.5f * s5 * (3.f * z * z - 1.f);
    Yp[7] = s15 * x * z;  Yp[8] = 0.5f * s15 * (x * x - y * y);
    float u  = fminf(fmaxf(rr * (1.f / RMAXF), 0.f), 1.f);
    float u2 = u * u, u6 = u2 * u2 * u2;
    float fcut = 1.f - 28.f * u6 + 48.f * u6 * u - 21.f * u6 * u2;
    float pref = sqrtf(2.f / RMAXF) * inv * fcut;
    float* Fp = &sF[tid * 12];
    #pragma unroll
    for (int nb = 0; nb < 8; ++nb)
      Fp[nb] = pref * __sinf((float)(nb + 1) * PI_F * rr * (1.f / RMAXF));
  }
  __syncthreads();

  const int lane = tid & 31, l16 = lane & 15, lh = lane >> 4;
  const int wv = tid >> 5;        // 0..7
  const int m0 = (wv & 1) * 16;   // row block (32 edges = 2 blocks)
  const int g  = wv >> 1;         // col group 0..3

  // ---- layer 1: [32,8] @ w1[8,64], silu -> sH1 ----
  {
    int n0 = g * 16;
    v8f acc = {};
    #pragma unroll
    for (int ks = 0; ks < 2; ++ks) {
      int k0 = ks * 4;
      v2f a, b;
      a.x = sF[(m0 + l16) * 12 + k0 + 2 * lh];
      a.y = sF[(m0 + l16) * 12 + k0 + 2 * lh + 1];
      b.x = w1[(k0 + 2 * lh) * 64 + n0 + l16];
      b.y = w1[(k0 + 2 * lh + 1) * 64 + n0 + l16];
      acc = __builtin_amdgcn_wmma_f32_16x16x4_f32(false, a, false, b,
                                                  (short)0, acc, false, false);
    }
    #pragma unroll
    for (int r8 = 0; r8 < 8; ++r8)
      sH1[(m0 + r8 + 8 * lh) * 68 + n0 + l16] = silu_fast(acc[r8]);
  }
  __syncthreads();

  // ---- layer 2: sH1 @ w2[64,64], silu -> sH2 ----
  {
    int n0 = g * 16;
    v8f acc = {};
    #pragma unroll
    for (int ks = 0; ks < 16; ++ks) {
      int k0 = ks * 4;
      v2f a, b;
      a.x = sH1[(m0 + l16) * 68 + k0 + 2 * lh];
      a.y = sH1[(m0 + l16) * 68 + k0 + 2 * lh + 1];
      b.x = w2[(k0 + 2 * lh) * 64 + n0 + l16];
      b.y = w2[(k0 + 2 * lh + 1) * 64 + n0 + l16];
      acc = __builtin_amdgcn_wmma_f32_16x16x4_f32(false, a, false, b,
                                                  (short)0, acc, false, false);
    }
    #pragma unroll
    for (int r8 = 0; r8 < 8; ++r8)
      sH2[(m0 + r8 + 8 * lh) * 68 + n0 + l16] = silu_fast(acc[r8]);
  }
  __syncthreads();

  // ---- layer 3: sH2 @ w3[64,64], silu -> sH1 ----
  {
    int n0 = g * 16;
    v8f acc = {};
    #pragma unroll
    for (int ks = 0; ks < 16; ++ks) {
      int k0 = ks * 4;
      v2f a, b;
      a.x = sH2[(m0 + l16) * 68 + k0 + 2 * lh];
      a.y = sH2[(m0 + l16) * 68 + k0 + 2 * lh + 1];
      b.x = w3[(k0 + 2 * lh) * 64 + n0 + l16];
      b.y = w3[(k0 + 2 * lh + 1) * 64 + n0 + l16];
      acc = __builtin_amdgcn_wmma_f32_16x16x4_f32(false, a, false, b,
                                                  (short)0, acc, false, false);
    }
    #pragma unroll
    for (int r8 = 0; r8 < 8; ++r8)
      sH1[(m0 + r8 + 8 * lh) * 68 + n0 + l16] = silu_fast(acc[r8]);
  }
  __syncthreads();

  // ---- layer 4: sH1 @ w4[64,192] -> sT (no activation) ----
  #pragma unroll
  for (int t = 0; t < 3; ++t) {
    int n0 = g * 48 + t * 16;
    v8f acc = {};
    #pragma unroll
    for (int ks = 0; ks < 16; ++ks) {
      int k0 = ks * 4;
      v2f a, b;
      a.x = sH1[(m0 + l16) * 68 + k0 + 2 * lh];
      a.y = sH1[(m0 + l16) * 68 + k0 + 2 * lh + 1];
      b.x = w4[(k0 + 2 * lh) * 192 + n0 + l16];
      b.y = w4[(k0 + 2 * lh + 1) * 192 + n0 + l16];
      acc = __builtin_amdgcn_wmma_f32_16x16x4_f32(false, a, false, b,
                                                  (short)0, acc, false, false);
    }
    #pragma unroll
    for (int r8 = 0; r8 < 8; ++r8)
      sT[(m0 + r8 + 8 * lh) * 196 + n0 + l16] = acc[r8];
  }
  __syncthreads();

  // ---- scatter: msg[recv, lm, c] += h_s[c] * tpw[e, L(lm), c] * Y[e, lm] / AVG
  const int c = tid & 63, sub = tid >> 6;   // 4 edges in flight, coalesced in c
  const int LOF[9] = {0, 1, 1, 1, 2, 2, 2, 2, 2};
  for (int eb = 0; eb < 8; ++eb) {
    int e = eb * 4 + sub;
    float hs = Hz[spec[sS[e]] * 64 + c] * (1.f / AVGF);
    const float* Yp = &sY[e * 12];
    const float* Tp = &sT[e * 196];
    float* mrow = &msg[(long)sR[e] * 576];
    #pragma unroll
    for (int lm = 0; lm < 9; ++lm) {
      float v = hs * Tp[LOF[lm] * 64 + c] * Yp[lm];
      atomicAdd(&mrow[lm * 64 + c], v);
    }
  }
}

// ---------------- per-node equivariant contraction, WMMA-accelerated --------
// res(n,c) = sum_{ijkp} U3[i,j,k,p] x_i x_j x_k W3[p]
//          + sum_{jkp}  U2[j,k,p]  x_j x_k      W2[p]
//          + sum_{jp}   U1[j,p]    x_j          W1[p]
// Stage 1 (WMMA): T2[row, i*4+p] = sum_jk X2[row,jk] * U3[i, jk, p]
//   with A = X2 [64 rows, K=81 pad 84], B = U3 flat view [84, 48] (i*4+p cols)
// One node per 128-thread block (rows = channels c=0..63; 4 waves x 16 rows).
__global__ __launch_bounds__(128) void node_kernel(
    const float* __restrict__ msg, const int* __restrict__ spec,
    const float* __restrict__ U3, const float* __restrict__ U2,
    const float* __restrict__ U1, const float* __restrict__ Wc3,
    const float* __restrict__ Wc2, const float* __restrict__ Wc1,
    float* __restrict__ tmp) {
  __shared__ float sB[84 * 48];    // U3 as B matrix [K=84][N=48], zero-padded
  __shared__ float sX2[64 * 84];   // A matrix: outer products x_j*x_k (ld 84)
  __shared__ float sT2[64 * 52];   // GEMM result [64 rows][48] (ld 52)
  __shared__ float sU2[9 * 9 * 4];
  __shared__ float sU1[9 * 4];

  const int tid = threadIdx.x;
  const int n   = blockIdx.x;

  // ---- build B = U3 reshaped [jk, i*4+p], zero pad jk>=81 ----
  for (int idx = tid; idx < 84 * 48; idx += 128) {
    int jk = idx / 48, col = idx % 48;
    int i = col >> 2, p = col & 3;
    sB[idx] = (jk < 81 && i < 9) ? U3[i * 324 + jk * 4 + p] : 0.f;
  }
  for (int i = tid; i < 324; i += 128) sU2[i] = U2[i];
  if (tid < 36) sU1[tid] = U1[tid];

  // ---- per-row x load + outer-product A rows ----
  float x[9];
  if (tid < 64) {
    const int c = tid;
    const float* mrow = &msg[(long)n * 576];
    #pragma unroll
    for (int i = 0; i < 9; ++i) x[i] = mrow[i * 64 + c];
    float* Xr = &sX2[c * 84];
    #pragma unroll
    for (int j = 0; j < 9; ++j)
      #pragma unroll
      for (int k = 0; k < 9; ++k) Xr[j * 9 + k] = x[j] * x[k];
    Xr[81] = 0.f; Xr[82] = 0.f; Xr[83] = 0.f;   // K padding
  }
  __syncthreads();

  // ---- stage 1: T2 = X2 @ B via WMMA f32 16x16x4 ----
  const int lane = tid & 31, l16 = lane & 15, lh = lane >> 4;
  const int m0 = (tid >> 5) * 16;   // 4 waves x 16 rows
  #pragma unroll
  for (int t = 0; t < 3; ++t) {
    int n0 = t * 16;
    v8f acc = {};
    #pragma unroll
    for (int ks = 0; ks < 21; ++ks) {
      int k0 = ks * 4;
      v2f a, b;
      a.x = sX2[(m0 + l16) * 84 + k0 + 2 * lh];
      a.y = sX2[(m0 + l16) * 84 + k0 + 2 * lh + 1];
      b.x = sB[(k0 + 2 * lh) * 48 + n0 + l16];
      b.y = sB[(k0 + 2 * lh + 1) * 48 + n0 + l16];
      acc = __builtin_amdgcn_wmma_f32_16x16x4_f32(false, a, false, b,
                                                  (short)0, acc, false, false);
    }
    #pragma unroll
    for (int r8 = 0; r8 < 8; ++r8)
      sT2[(m0 + r8 + 8 * lh) * 52 + n0 + l16] = acc[r8];
  }
  __syncthreads();

  // ---- stage 2: per-row reduction with species-dependent weights ----
  if (tid < 64) {
    const int c = tid;
    const int z = spec[n];
    float W3[4], W2[4], W1[4];
    #pragma unroll
    for (int p = 0; p < 4; ++p) {
      W3[p] = Wc3[(z * 4 + p) * 64 + c];
      W2[p] = Wc2[(z * 4 + p) * 64 + c];
      W1[p] = Wc1[(z * 4 + p) * 64 + c];
    }
    float res = 0.f;
    const float* Tr = &sT2[c * 52];
    #pragma unroll
    for (int i = 0; i < 9; ++i) {
      float s = 0.f;
      #pragma unroll
      for (int p = 0; p < 4; ++p) s += Tr[i * 4 + p] * W3[p];
      res += s * x[i];
    }
    const float* Xr = &sX2[c * 84];
    #pragma unroll
    for (int p = 0; p < 4; ++p) {
      float s = 0.f;
      for (int jk = 0; jk < 81; ++jk) s += sU2[jk * 4 + p] * Xr[jk];
      res += s * W2[p];
    }
    #pragma unroll
    for (int p = 0; p < 4; ++p) {
      float s = 0.f;
      #pragma unroll
      for (int j = 0; j < 9; ++j) s += sU1[j * 4 + p] * x[j];
      res += s * W1[p];
    }
    tmp[(long)n * 64 + c] = res;
  }
}

// ---------------- final GEMM: out = tmp[N,64] @ W_out[64,64] / 8 (WMMA) -----
__global__ __launch_bounds__(128) void out_kernel(
    const float* __restrict__ tmp, const float* __restrict__ Wout,
    float* __restrict__ out) {
  __shared__ float sA[64 * 68];
  const int tid = threadIdx.x;
  const int nbase = blockIdx.x * 64;
  for (int idx = tid; idx < 64 * 64; idx += 128) {
    int rr = idx >> 6, cc = idx & 63;
    int n = nbase + rr;
    sA[rr * 68 + cc] = (n < NN) ? tmp[(long)n * 64 + cc] : 0.f;
  }
  __syncthreads();

  const int lane = tid & 31, l16 = lane & 15, lh = lane >> 4;
  const int m0 = (tid >> 5) * 16;   // 4 waves x 16 rows
  #pragma unroll
  for (int t = 0; t < 4; ++t) {
    int n0 = t * 16;
    v8f acc = {};
    #pragma unroll
    for (int ks = 0; ks < 16; ++ks) {
      int k0 = ks * 4;
      v2f a, b;
      a.x = sA[(m0 + l16) * 68 + k0 + 2 * lh];
      a.y = sA[(m0 + l16) * 68 + k0 + 2 * lh + 1];
      b.x = Wout[(k0 + 2 * lh) * 64 + n0 + l16];
      b.y = Wout[(k0 + 2 * lh + 1) * 64 + n0 + l16];
      acc = __builtin_amdgcn_wmma_f32_16x16x4_f32(false, a, false, b,
                                                  (short)0, acc, false, false);
    }
    #pragma unroll
    for (int r8 = 0; r8 < 8; ++r8) {
      int n = nbase + m0 + r8 + 8 * lh;
      if (n < NN) out[(long)n * 64 + n0 + l16] = acc[r8] * 0.125f;
    }
  }
}

// ---------------- host launcher --------------------------------------------
extern "C" void kernel_launch(void* const* d_in, const int* in_sizes, int n_in,
                              void* d_out, int out_size, void* d_ws, size_t ws_size,
                              hipStream_t stream) {
  const float* positions = (const float*)d_in[0];
  const int*   spec      = (const int*)d_in[1];
  const int*   eidx      = (const int*)d_in[2];
  const float* W_embed   = (const float*)d_in[3];
  const float* W_up      = (const float*)d_in[4];
  const float* mlp_w1    = (const float*)d_in[5];
  const float* mlp_w2    = (const float*)d_in[6];
  const float* mlp_w3    = (const float*)d_in[7];
  const float* mlp_w4    = (const float*)d_in[8];
  const float* U3        = (const float*)d_in[9];
  const float* U2        = (const float*)d_in[10];
  const float* U1        = (const float*)d_in[11];
  const float* Wc3       = (const float*)d_in[12];
  const float* Wc2       = (const float*)d_in[13];
  const float* Wc1       = (const float*)d_in[14];
  const float* W_out     = (const float*)d_in[15];
  float* out = (float*)d_out;

  char* ws = (char*)d_ws;
  float* Hz  = (float*)ws;                                  // 640 floats
  float* msg = (float*)(ws + 4096);                         // N*9*64 floats
  float* tmp = (float*)(ws + 4096 + (size_t)NN * 576 * 4);  // N*64 floats

  hipLaunchKernelGGL(hz_kernel, dim3(ZZ), dim3(CC), 0, stream, W_embed, W_up, Hz);
  int n4 = (NN * 576) / 4;
  hipLaunchKernelGGL(zero_kernel, dim3((n4 + 255) / 256), dim3(256), 0, stream,
                     (float4*)msg, n4);
  hipLaunchKernelGGL(edge_kernel, dim3(EE / 32), dim3(256), 0, stream,
                     positions, spec, eidx, Hz, mlp_w1, mlp_w2, mlp_w3, mlp_w4, msg);
  hipLaunchKernelGGL(node_kernel, dim3(NN), dim3(128), 0, stream,
                     msg, spec, U3, U2, U1, Wc3, Wc2, Wc1, tmp);
  hipLaunchKernelGGL(out_kernel, dim3((NN + 63) / 64), dim3(128), 0, stream,
                     tmp, W_out, out);
}